// RowColCrossTransformer_69758858821957
// MI455X (gfx1250) — compile-verified
//
#include <hip/hip_runtime.h>
#include <hip/hip_bf16.h>
#include <math.h>

typedef __attribute__((ext_vector_type(2))) float v2f;
typedef __attribute__((ext_vector_type(8))) float v8f;

#define NFIELD 65
#define NF1    64          // nfield - 1 (attention columns)
#define NEMB   256
#define NHEAD  8
#define HD     32
#define KSTRIDE 260        // padded LDS row stride (floats), 16B aligned, conflict-breaking
#define ROW_IN  (NFIELD * NEMB)            // 16640 floats per input row
#define ROW_OUT ((NFIELD + NHEAD) * NEMB)  // 18688 floats per output row
#define N_ITER 50

// ---------------------------------------------------------------------------
// Tiny precompute: w2[h*32+d] = scale * sum_k bilinear_w[h,d,k] * query[h,k]
// scale = nemb^-0.5 = 1/16. 64K MACs, one block.
// ---------------------------------------------------------------------------
__global__ __launch_bounds__(256) void precompute_w2_kernel(
    const float* __restrict__ bw, const float* __restrict__ q,
    float* __restrict__ w2) {
  int t = threadIdx.x;              // t -> (h, d)
  int h = t >> 5, d = t & 31;
  const float* bwp = bw + ((size_t)h * HD + d) * NEMB;
  const float* qp  = q  + (size_t)h * NEMB;
  float acc = 0.f;
  #pragma unroll 8
  for (int k = 0; k < NEMB; ++k) acc += bwp[k] * qp[k];
  w2[t] = acc * 0.0625f;
}

// ---------------------------------------------------------------------------
// Main fused kernel: one block (8 wave32) per row r.
// ---------------------------------------------------------------------------
__global__ __launch_bounds__(256) void entmax_cross_kernel(
    const float* __restrict__ x, const float* __restrict__ values,
    const float* __restrict__ w2g, float* __restrict__ out) {
  __shared__ float Klds[NF1 * KSTRIDE];   // K = x[r,1:65,:]; reused for x_cross staging
  __shared__ float cwlds[16 * NF1];       // WMMA A-panel, heads padded 8->16
  __shared__ float w2lds[NHEAD * HD];

  const int tid  = threadIdx.x;
  const int wave = tid >> 5;              // 8 waves == 8 heads
  const int lane = tid & 31;
  const int r    = blockIdx.x;

  const float* xrow = x   + (size_t)r * ROW_IN;
  float*       orow = out + (size_t)r * ROW_OUT;

  w2lds[tid] = w2g[tid];

  // ---- pass-through copy of x row + scatter fields 1..64 into LDS ----
  const float4* xv = (const float4*)xrow;
  float4*       ov = (float4*)orow;
  #pragma unroll 4
  for (int j = tid; j < ROW_IN / 4; j += 256) {
    float4 v = xv[j];
    ov[j] = v;
    if (j >= NEMB / 4) {                  // skip field 0
      int c = (j >> 6) - 1;               // field index 0..63
      int e = (j & 63) << 2;              // embedding offset
      *(float4*)&Klds[c * KSTRIDE + e] = v;
    }
  }
  // zero padded A-panel rows 8..15
  cwlds[512 + tid] = 0.f;
  cwlds[768 + tid] = 0.f;
  __syncthreads();

  // ---- att[h=wave, c] for c = lane and lane+32 ----
  const int c0 = lane, c1 = lane + 32;
  const float* wrow = &w2lds[wave * HD];
  const float* kp0  = &Klds[c0 * KSTRIDE + wave * HD];
  const float* kp1  = &Klds[c1 * KSTRIDE + wave * HD];
  float a0 = 0.f, a1 = 0.f;
  #pragma unroll
  for (int d = 0; d < HD; ++d) {
    float wv = wrow[d];                   // broadcast LDS read
    a0 += kp0[d] * wv;
    a1 += kp1[d] * wv;
  }

  // ---- 1.5-entmax bisection (Xs = att*0.5, p = relu(Xs-tau)^2) ----
  float xs0 = a0 * 0.5f, xs1 = a1 * 0.5f;
  float m = fmaxf(xs0, xs1);
  #pragma unroll
  for (int off = 16; off; off >>= 1) m = fmaxf(m, __shfl_xor(m, off, 32));
  float tau_lo = m - 1.0f;
  float dm = 0.875f;                      // tau_hi - tau_lo = 1 - (1/64)^0.5
  float t0 = fmaxf(xs0 - tau_lo, 0.f), t1 = fmaxf(xs1 - tau_lo, 0.f);
  float s = t0 * t0 + t1 * t1;
  #pragma unroll
  for (int off = 16; off; off >>= 1) s += __shfl_xor(s, off, 32);
  const float f_lo = s - 1.f;             // fixed, as in reference
  float tau_m = tau_lo;
  for (int it = 0; it < N_ITER; ++it) {
    dm *= 0.5f;
    tau_m = tau_lo + dm;
    t0 = fmaxf(xs0 - tau_m, 0.f); t1 = fmaxf(xs1 - tau_m, 0.f);
    s = t0 * t0 + t1 * t1;
    #pragma unroll
    for (int off = 16; off; off >>= 1) s += __shfl_xor(s, off, 32);
    float f_m = s - 1.f;
    if (f_m * f_lo >= 0.f) tau_lo = tau_m;   // wave-uniform branch
  }
  t0 = fmaxf(xs0 - tau_m, 0.f); t1 = fmaxf(xs1 - tau_m, 0.f);
  float p0 = t0 * t0, p1 = t1 * t1;
  s = p0 + p1;
  #pragma unroll
  for (int off = 16; off; off >>= 1) s += __shfl_xor(s, off, 32);
  float inv = 1.f / s;
  const float* vrow = values + wave * NF1;
  cwlds[wave * NF1 + c0] = p0 * inv * vrow[c0];
  cwlds[wave * NF1 + c1] = p1 * inv * vrow[c1];
  __syncthreads();

  // ---- D[16x256] = CW[16x64] x K[64x256] via v_wmma_f32_16x16x4_f32 ----
  // A 16x4 f32 layout: lanes 0-15 -> M=lane, K={k0,k0+1}; lanes 16-31 -> K={k0+2,k0+3}
  // B 4x16  f32 layout mirrored: lanes 0-15 -> K={k0,k0+1}, N=lane; lanes 16-31 -> K={k0+2,k0+3}
  const int mrow  = lane & 15;
  const int khalf = (lane >> 4) << 1;
  const int n0 = wave * 32, n1 = n0 + 16;
  const int ncol0 = n0 + (lane & 15), ncol1 = n1 + (lane & 15);
  v8f acc0 = {}, acc1 = {};
  #pragma unroll
  for (int kc = 0; kc < 16; ++kc) {
    int ks = 4 * kc + khalf;
    v2f A, B0, B1;
    A.x  = cwlds[mrow * NF1 + ks];
    A.y  = cwlds[mrow * NF1 + ks + 1];
    B0.x = Klds[ks * KSTRIDE + ncol0];
    B0.y = Klds[(ks + 1) * KSTRIDE + ncol0];
    B1.x = Klds[ks * KSTRIDE + ncol1];
    B1.y = Klds[(ks + 1) * KSTRIDE + ncol1];
    acc0 = __builtin_amdgcn_wmma_f32_16x16x4_f32(
        false, A, false, B0, (short)0, acc0, false, false);
    acc1 = __builtin_amdgcn_wmma_f32_16x16x4_f32(
        false, A, false, B1, (short)0, acc1, false, false);
  }
  __syncthreads();                        // all K/cw LDS reads complete

  // ---- exp(D) staged into LDS (reuse Klds), heads 0..7 in lanes 0..15 ----
  float* xc = Klds;
  if (lane < 16) {
    #pragma unroll
    for (int j = 0; j < 8; ++j) {        // accumulator VGPR j -> head j (lanes 0-15)
      xc[j * NEMB + n0 + lane] = expf(acc0[j]);
      xc[j * NEMB + n1 + lane] = expf(acc1[j]);
    }
  }
  __syncthreads();

  // ---- coalesced store of x_cross (2048 floats = 512 float4) ----
  float4* xcv = (float4*)xc;
  float4* oxv = (float4*)(orow + ROW_IN);
  oxv[tid]        = xcv[tid];
  oxv[tid + 256]  = xcv[tid + 256];
}

// ---------------------------------------------------------------------------
extern "C" void kernel_launch(void* const* d_in, const int* in_sizes, int n_in,
                              void* d_out, int out_size, void* d_ws, size_t ws_size,
                              hipStream_t stream) {
  const float* x    = (const float*)d_in[0];   // [1, R, 65*256]
  const float* bw   = (const float*)d_in[1];   // [8, 32, 256]
  const float* q    = (const float*)d_in[2];   // [8, 256]
  const float* vals = (const float*)d_in[3];   // [8, 64]
  float* out = (float*)d_out;
  float* w2  = (float*)d_ws;                   // 256 floats of scratch

  int R = in_sizes[0] / ROW_IN;

  precompute_w2_kernel<<<1, 256, 0, stream>>>(bw, q, w2);
  entmax_cross_kernel<<<R, 256, 0, stream>>>(x, vals, w2, out);
}